// MaxPooling_31739808317550
// MI455X (gfx1250) — compile-verified
//
#include <hip/hip_runtime.h>
#include <cstdint>
#include <cstddef>

#define CCH 64  // feature channels

// ---------------------------------------------------------------------------
// Kernel 1: fill output with -inf (segment_max identity), float4 stores.
// ---------------------------------------------------------------------------
__global__ __launch_bounds__(256) void mp_init_kernel(float4* __restrict__ out, int n4) {
    int i = blockIdx.x * blockDim.x + threadIdx.x;
    if (i >= n4) return;
    const float ninf = -__builtin_inff();
    out[i] = make_float4(ninf, ninf, ninf, ninf);
}

// ---------------------------------------------------------------------------
// Kernel 2: masked gather + scatter-max via native VMEM float atomics.
// One wave32 owns 32 edges:
//   - lane l loads metadata for edge (base + l) coalesced
//   - __ballot the validity mask (wave-uniform) and bit-scan live edges
//   - v_readlane broadcasts src/tgt into SGPRs -> scalar row bases (SADDR form)
//   - 32 lanes cover the 64-channel row as float2 each (256B coalesced b64)
//   - two global_atomic_max_num_f32 per lane (offset:0 / offset:4), DEV scope
// No "memory" clobber: nothing here reads `out`, so independent gathers may be
// scheduled past in-flight atomics for MLP.
// ---------------------------------------------------------------------------
__global__ __launch_bounds__(256) void mp_scatter_kernel(
    const float* __restrict__ feat,
    const int*   __restrict__ src_ids,
    const int*   __restrict__ tgt_ids,
    const int*   __restrict__ ntypes,
    float*       __restrict__ out,
    int num_edges)
{
    const int lane = threadIdx.x & 31;
    const int wave = blockIdx.x * (blockDim.x >> 5) + (threadIdx.x >> 5);
    const int base = wave << 5;              // 32 edges per wave
    if (base >= num_edges) return;

    int ok = 0, src = 0, tgt = 0;
    const int e = base + lane;
    if (e < num_edges) {
        // sign bit of OR == OR of sign bits -> all >= 0 iff OR >= 0
        const int t0 = ntypes[3 * e + 0];
        const int t1 = ntypes[3 * e + 1];
        const int t2 = ntypes[3 * e + 2];
        ok  = ((t0 | t1 | t2) >= 0) ? 1 : 0;
        src = src_ids[e];
        tgt = tgt_ids[e];
    }

    unsigned live = (unsigned)__ballot(ok);          // wave32: low 32 bits, uniform
    const unsigned lane8 = (unsigned)(lane << 3);    // byte offset of this lane's float2

    while (live) {
        const int j = __builtin_ctz(live);
        live &= live - 1u;

        // Uniform broadcasts -> SGPRs -> scalar row bases
        const int s = __builtin_amdgcn_readlane(src, j);
        const int t = __builtin_amdgcn_readlane(tgt, j);

        const float2 v = *(const float2*)((const char*)(feat + (size_t)t * CCH) + lane8);
        float* prow = out + (size_t)s * CCH;

        // Native CDNA5 float atomic max, SADDR form: s[row] + v_lane8 (+imm)
        asm volatile("global_atomic_max_num_f32 %0, %1, %2 scope:SCOPE_DEV"
                     :: "v"(lane8), "v"(v.x), "s"(prow));
        asm volatile("global_atomic_max_num_f32 %0, %1, %2 offset:4 scope:SCOPE_DEV"
                     :: "v"(lane8), "v"(v.y), "s"(prow));
    }
}

// ---------------------------------------------------------------------------
// Kernel 3: empty segments were left at -inf -> set to 0 (in place, float4).
// ---------------------------------------------------------------------------
__global__ __launch_bounds__(256) void mp_finalize_kernel(float4* __restrict__ out, int n4) {
    int i = blockIdx.x * blockDim.x + threadIdx.x;
    if (i >= n4) return;
    const float ninf = -__builtin_inff();
    float4 v = out[i];
    v.x = (v.x == ninf) ? 0.0f : v.x;
    v.y = (v.y == ninf) ? 0.0f : v.y;
    v.z = (v.z == ninf) ? 0.0f : v.z;
    v.w = (v.w == ninf) ? 0.0f : v.w;
    out[i] = v;
}

// ---------------------------------------------------------------------------
// Kernel 4: tuple's second output (feat_depth + 1) if appended to d_out.
// ---------------------------------------------------------------------------
__global__ void mp_tail_kernel(float* __restrict__ dst,
                               const int* __restrict__ feat_depth, int n) {
    int i = blockIdx.x * blockDim.x + threadIdx.x;
    if (i < n) dst[i] = (float)(feat_depth[0] + 1);
}

extern "C" void kernel_launch(void* const* d_in, const int* in_sizes, int n_in,
                              void* d_out, int out_size, void* d_ws, size_t ws_size,
                              hipStream_t stream) {
    const float* feat       = (const float*)d_in[0];
    const int*   src_ids    = (const int*)  d_in[1];
    const int*   tgt_ids    = (const int*)  d_in[2];
    const int*   ntypes     = (const int*)  d_in[3];
    // d_in[4] = dim_size (implied by out_size), d_in[5] = feat_depth
    const int*   feat_depth = (n_in > 5) ? (const int*)d_in[5] : nullptr;
    float*       out        = (float*)d_out;

    const int E      = in_sizes[1];            // number of edges (src_ids length)
    const int n_out  = out_size / CCH;         // pooled rows
    const int total  = n_out * CCH;            // matrix elements in d_out
    const int tail   = out_size - total;       // 0 or 1 (appended scalar)
    const int n4     = total / 4;

    // 1) init to -inf
    mp_init_kernel<<<(n4 + 255) / 256, 256, 0, stream>>>((float4*)out, n4);

    // 2) scatter-max: 32 edges per wave, 8 waves per block -> 256 edges/block
    const int edges_per_block = 256;           // (256 threads / 32) * 32 edges
    const int nblocks = (E + edges_per_block - 1) / edges_per_block;
    mp_scatter_kernel<<<nblocks, 256, 0, stream>>>(feat, src_ids, tgt_ids, ntypes,
                                                   out, E);

    // 3) -inf -> 0 for empty segments
    mp_finalize_kernel<<<(n4 + 255) / 256, 256, 0, stream>>>((float4*)out, n4);

    // 4) optional appended scalar output
    if (tail > 0 && feat_depth != nullptr) {
        mp_tail_kernel<<<1, 32, 0, stream>>>(out + total, feat_depth, tail);
    }
}